// CausalGraphModule_10350871183765
// MI455X (gfx1250) — compile-verified
//
#include <hip/hip_runtime.h>

// ---------------------------------------------------------------------------
// CausalGraphModule for MI455X (gfx1250, wave32)
//
// Cost model:
//   * scan of state_sequence: 128 MiB HBM (L2 holds the whole tensor for the
//     second touch) -> ~6 us at 23.3 TB/s. Bandwidth bound; b128 loads.
//   * pair MLP: 1024x256 @ 256x128 f32 GEMM = 67 MFLOP -> V_WMMA_F32_16X16X4_F32
//     (exact f32 math, matches reference numerics).
//   * outputs: 4 KiB sigmoid(cm) + 1.5 MB preds broadcast. Negligible.
// ---------------------------------------------------------------------------

typedef __attribute__((ext_vector_type(2))) float v2f;
typedef __attribute__((ext_vector_type(8))) float v8f;

#define N_OBJ   32
#define D_MODEL 128
#define T_STEPS 127
#define HW      (128 * 128)
#define N_PAIRS 992            // 32*31 off-diagonal
#define PRED_ELTS (T_STEPS * N_PAIRS * 3)

// ws layout (uint32 units):
//   [0   .. 126]  presMask[t] : bit i set iff object i present in prev frame t
//   [127 .. 253]  flags[t]    : bit0 = any cell equal (has0), bit1 = any diff (has1)
//   [256 .. ]     pair_mech[1024*3] floats (dense i-major NxN mech outputs)

__global__ void init_ws_kernel(unsigned* __restrict__ ws) {
    if (threadIdx.x < 2 * T_STEPS) ws[threadIdx.x] = 0u;
}

// grid (127, 16, 4), block 256. Each block: 4096 cells of one (b, t) slice.
__global__ void scan_frames_kernel(const float* __restrict__ state,
                                   unsigned* __restrict__ presMask,
                                   unsigned* __restrict__ flags) {
    const int t = blockIdx.x;          // 0..126 (prev frame index)
    const int b = blockIdx.y;          // 0..15
    const int chunk = blockIdx.z;      // 0..3

    const float* __restrict__ prevF = state + ((size_t)b * 128 + (size_t)t) * (size_t)HW;
    const float* __restrict__ currF = prevF + HW;

    unsigned m = 0u, f = 0u;
#pragma unroll
    for (int it = 0; it < 4; ++it) {
        const int idx = chunk * 4096 + (it * 256 + (int)threadIdx.x) * 4;
        const float4 p = *(const float4*)(prevF + idx);   // global_load_b128
        const float4 c = *(const float4*)(currF + idx);   // global_load_b128
        m |= (1u << (int)p.x) | (1u << (int)p.y) | (1u << (int)p.z) | (1u << (int)p.w);
        f |= (p.x == c.x ? 1u : 2u) | (p.y == c.y ? 1u : 2u) |
             (p.z == c.z ? 1u : 2u) | (p.w == c.w ? 1u : 2u);
    }

    __shared__ unsigned sm, sf;
    if (threadIdx.x == 0) { sm = 0u; sf = 0u; }
    __syncthreads();
    atomicOr(&sm, m);
    atomicOr(&sf, f);
    __syncthreads();
    if (threadIdx.x == 0) {
        atomicOr(&presMask[t], sm);
        atomicOr(&flags[t], sf);
    }
}

// grid 64 blocks (16 pair-rows each), block 256 = 8 waves.
// Wave w computes hidden columns [16w, 16w+16) via V_WMMA_F32_16X16X4_F32,
// then folds relu(h+b1) @ W2 into LDS with ds_add_f32.
__global__ void pair_mlp_wmma_kernel(const float* __restrict__ emb,
                                     const float* __restrict__ W1,
                                     const float* __restrict__ b1,
                                     const float* __restrict__ W2,
                                     const float* __restrict__ b2,
                                     float* __restrict__ pair_mech) {
    __shared__ float smech[16][3];
    const int tid = threadIdx.x;
    if (tid < 48) ((float*)smech)[tid] = 0.0f;
    __syncthreads();

    const int wave = tid >> 5;        // 0..7 -> N-tile
    const int lane = tid & 31;
    const int half = lane >> 4;       // 0 or 1 (lane group)
    const int l16  = lane & 15;

    // A fragment: row = pair p; comb[p][k] = k<128 ? emb[i][k] : emb[j][k-128]
    const int p = blockIdx.x * 16 + l16;
    const int i = p >> 5, j = p & 31;
    const float* __restrict__ rowL = emb + (size_t)i * D_MODEL;
    const float* __restrict__ rowR = emb + (size_t)j * D_MODEL;

    // B fragment: column n of W1 for this lane
    const int n = wave * 16 + l16;

    v8f acc = {};
    for (int ks = 0; ks < 64; ++ks) {
        // f32 16x4 A layout: VGPR0 = K {0|2} by lane-half, VGPR1 = K {1|3}.
        const int k0 = ks * 4 + half * 2;          // even; k0,k0+1 same half of comb
        const float* __restrict__ src = (k0 < D_MODEL) ? (rowL + k0)
                                                       : (rowR + (k0 - D_MODEL));
        v2f a, bm;
        a.x = src[0];
        a.y = src[1];
        bm.x = W1[(size_t)k0 * D_MODEL + n];
        bm.y = W1[(size_t)(k0 + 1) * D_MODEL + n];
        acc = __builtin_amdgcn_wmma_f32_16x16x4_f32(
            /*neg_a=*/false, a, /*neg_b=*/false, bm,
            /*c_mod=*/(short)0, acc, /*reuse_a=*/false, /*reuse_b=*/false);
    }

    // acc[r] = hidden[M = r + 8*half][N = n]; apply bias+relu, fold through W2.
    const float bias = b1[n];
    const float w2_0 = W2[n * 3 + 0];
    const float w2_1 = W2[n * 3 + 1];
    const float w2_2 = W2[n * 3 + 2];
#pragma unroll
    for (int r = 0; r < 8; ++r) {
        float h = acc[r] + bias;
        h = h > 0.0f ? h : 0.0f;
        const int m = r + 8 * half;
        atomicAdd(&smech[m][0], h * w2_0);   // ds_add_f32
        atomicAdd(&smech[m][1], h * w2_1);
        atomicAdd(&smech[m][2], h * w2_2);
    }
    __syncthreads();
    if (tid < 48) {
        const int m = tid / 3, o = tid % 3;
        pair_mech[(size_t)(blockIdx.x * 16 + m) * 3 + o] = smech[m][o] + b2[o];
    }
}

// 1 block, 1024 threads: out[i*32+j] = sigmoid(causal + 0.01*count - diag mask)
__global__ void cm_sigmoid_kernel(const unsigned* __restrict__ presMask,
                                  const unsigned* __restrict__ flags,
                                  const float* __restrict__ causal,
                                  float* __restrict__ out) {
    __shared__ unsigned sp[T_STEPS], sf[T_STEPS];
    const int tid = threadIdx.x;
    if (tid < T_STEPS) { sp[tid] = presMask[tid]; sf[tid] = flags[tid]; }
    __syncthreads();

    const int i = tid >> 5, j = tid & 31;
    int cnt = 0;
    if (j < 2) {
        const unsigned jb = 1u << j;
        for (int t = 0; t < T_STEPS; ++t)
            cnt += (int)(((sp[t] >> i) & 1u) & ((sf[t] & jb) ? 1u : 0u));
    }
    const float inc = (i == j) ? 0.0f : 0.01f * (float)cnt;
    const float cm = causal[tid] + inc;
    out[tid] = 1.0f / (1.0f + expf(-cm));
}

// preds: t-major broadcast of the 992 off-diagonal mech rows.
__global__ void write_preds_kernel(const float* __restrict__ pair_mech,
                                   float* __restrict__ out) {
    const int e = blockIdx.x * blockDim.x + threadIdx.x;
    if (e >= PRED_ELTS) return;
    const int row = e / 3, o = e - row * 3;
    const int p_off = row % N_PAIRS;          // pair index within timestep
    const int i = p_off / 31;
    const int r = p_off - i * 31;
    const int j = r + (r >= i ? 1 : 0);       // skip diagonal, i-major order
    out[N_OBJ * N_OBJ + e] = pair_mech[(size_t)(i * 32 + j) * 3 + o];
}

extern "C" void kernel_launch(void* const* d_in, const int* in_sizes, int n_in,
                              void* d_out, int out_size, void* d_ws, size_t ws_size,
                              hipStream_t stream) {
    const float* state  = (const float*)d_in[0];
    const float* emb    = (const float*)d_in[1];
    const float* W1     = (const float*)d_in[2];
    const float* b1     = (const float*)d_in[3];
    const float* W2     = (const float*)d_in[4];
    const float* b2     = (const float*)d_in[5];
    const float* causal = (const float*)d_in[6];
    float* out = (float*)d_out;

    unsigned* ws       = (unsigned*)d_ws;
    unsigned* presMask = ws;              // [127]
    unsigned* flags    = ws + T_STEPS;    // [127]
    float*    pair_mech = (float*)(ws + 256);   // [1024*3]

    hipLaunchKernelGGL(init_ws_kernel, dim3(1), dim3(256), 0, stream, ws);
    hipLaunchKernelGGL(scan_frames_kernel, dim3(T_STEPS, 16, 4), dim3(256), 0, stream,
                       state, presMask, flags);
    hipLaunchKernelGGL(pair_mlp_wmma_kernel, dim3(64), dim3(256), 0, stream,
                       emb, W1, b1, W2, b2, pair_mech);
    hipLaunchKernelGGL(cm_sigmoid_kernel, dim3(1), dim3(1024), 0, stream,
                       presMask, flags, causal, out);
    hipLaunchKernelGGL(write_preds_kernel, dim3((PRED_ELTS + 255) / 256), dim3(256),
                       0, stream, pair_mech, out);
}